// AvgReadout_18966575579290
// MI455X (gfx1250) — compile-verified
//
#include <hip/hip_runtime.h>
#include <stdint.h>

// ---------------------------------------------------------------------------
// AvgReadout: COO segment-mean (1.6M edges -> 100K nodes, d=128) + L2 norm.
// CDNA5 plan: wave32 owns a 128-f32 row (32 lanes x float4); async
// global->LDS B128 gathers pipelined 4-deep per wave (ASYNCcnt); no-return
// global f32 atomics for the segment sum; shfl_xor wave reduction for norm.
// cdna5_probe (never launched) is first in the file purely so the disasm
// snippet shows the exact lowering of the async load / wait / atomic path.
// ---------------------------------------------------------------------------

#define WAVES_PER_BLOCK 8
#define PIPE 4   // async pipeline depth per wave (power of two)

typedef int v4i __attribute__((ext_vector_type(4)));
typedef __attribute__((address_space(1))) v4i g_v4i;   // global 16B vector
typedef __attribute__((address_space(3))) v4i l_v4i;   // LDS 16B vector

__device__ __forceinline__ void async_copy_b128(const void* gp, void* lp) {
#if defined(__has_builtin) && __has_builtin(__builtin_amdgcn_global_load_async_to_lds_b128)
  __builtin_amdgcn_global_load_async_to_lds_b128(
      (g_v4i*)gp, (l_v4i*)lp, /*imm offset*/0, /*cpol*/0);
#else
  unsigned lds_off = (unsigned)(uintptr_t)lp;           // low 32b = LDS offset
  unsigned long long ga = (unsigned long long)(uintptr_t)gp;
  asm volatile("global_load_async_to_lds_b128 %0, %1, off"
               :: "v"(lds_off), "v"(ga) : "memory");
#endif
}

__device__ __forceinline__ void wait_async_le(void) {
#if defined(__has_builtin) && __has_builtin(__builtin_amdgcn_s_wait_asynccnt)
  __builtin_amdgcn_s_wait_asynccnt((unsigned short)(PIPE - 1));
#else
  asm volatile("s_wait_asynccnt %0" :: "i"(PIPE - 1));
#endif
}

// Hardware wait for DS data + compiler reordering barrier: guarantees the
// ds_load consuming a pipeline slot has completed before the async refill
// of that same slot is issued.
__device__ __forceinline__ void ds_drain_barrier(void) {
  asm volatile("s_wait_dscnt 0x0" ::: "memory");
}

__device__ __forceinline__ void atomic_add_f32(float* p, float v) {
  // relaxed, agent scope -> global_atomic_add_f32 (no return, STOREcnt)
  __hip_atomic_fetch_add(p, v, __ATOMIC_RELAXED, __HIP_MEMORY_SCOPE_AGENT);
}

// ---------------------------------------------------------------------------
// Opcode probe (NEVER launched): minimal body so the disasm snippet shows
// global_load_async_to_lds_b128 / s_wait_asynccnt / ds_load_b128 /
// global_atomic_add_f32 lowering directly.
// ---------------------------------------------------------------------------
__global__ void cdna5_probe(const float* __restrict__ g, float* __restrict__ out) {
  __shared__ float buf[32 * 4];
  const int t = threadIdx.x & 31;
  async_copy_b128(g + t * 4, &buf[t * 4]);
  wait_async_le();
  float4 v = *(const float4*)&buf[t * 4];
  atomic_add_f32(out + t, v.x + v.y + v.z + v.w);
}

// ---------------------------------------------------------------------------
// Edge scatter. One edge per wave per pipeline slot.
//   slot fill : each lane async-loads its 16B of emb[dst] into LDS
//   slot drain: wait asynccnt, ds_load float4, 4 global f32 atomics into vsum,
//               lane0 bumps deg[src]; then refill the slot with edge e+PIPE*W.
// Async loads complete in order per wave, so s_wait_asynccnt(PIPE-1) ==
// "oldest slot ready". Tail iterations issue clamped dummy loads (never
// consumed) to keep the outstanding count constant for the immediate wait.
// ---------------------------------------------------------------------------
__global__ void __launch_bounds__(WAVES_PER_BLOCK * 32)
scatter_kernel(const float* __restrict__ emb,
               const long long* __restrict__ mask,
               float* __restrict__ vsum,
               float* __restrict__ deg,
               int E, int totalWaves) {
  __shared__ float lds[WAVES_PER_BLOCK * PIPE * 128];
  if (E <= 0) return;

  const int lane = threadIdx.x & 31;
  const int wib  = threadIdx.x >> 5;
  const int gw   = blockIdx.x * WAVES_PER_BLOCK + wib;
  float* slot0 = &lds[wib * PIPE * 128];

  const long long* __restrict__ srcs = mask;      // mask[0,:]
  const long long* __restrict__ dsts = mask + E;  // mask[1,:]

  // Prologue: fill all PIPE slots (clamped if past the end).
  #pragma unroll
  for (int k = 0; k < PIPE; ++k) {
    long long e = (long long)gw + (long long)k * totalWaves;
    if (e >= E) e = E - 1;                       // harmless dummy
    long long d = dsts[e];
    const float* g = emb + d * 128 + lane * 4;
    float* l = slot0 + k * 128 + lane * 4;
    async_copy_b128(g, l);
  }

  int k = 0;
  for (long long e = gw; e < E; e += totalWaves, ++k) {
    long long src = srcs[e];

    wait_async_le();                             // oldest slot landed in LDS
    float* l = slot0 + (k & (PIPE - 1)) * 128 + lane * 4;
    float4 v = *(const float4*)l;                // ds_load_b128

    float* dp = vsum + src * 128 + lane * 4;
    atomic_add_f32(dp + 0, v.x);
    atomic_add_f32(dp + 1, v.y);
    atomic_add_f32(dp + 2, v.z);
    atomic_add_f32(dp + 3, v.w);
    if (lane == 0) atomic_add_f32(deg + src, 1.0f);

    // Slot data fully in VGPRs before the async engine may overwrite it.
    ds_drain_barrier();

    // Refill this slot with edge e + PIPE*stride (clamped dummy at tail).
    long long en = e + (long long)PIPE * totalWaves;
    if (en >= E) en = E - 1;
    long long dn = dsts[en];
    const float* g = emb + dn * 128 + lane * 4;
    async_copy_b128(g, l);
  }
  // Dangling clamped loads drain before S_ENDPGM (implicit wait-idle).
}

// ---------------------------------------------------------------------------
// Finalize. One wave per row: mean = vsum/max(deg,1), then L2-normalize
// with eps=1e-12 via wave32 shfl_xor reduction.
// ---------------------------------------------------------------------------
__global__ void __launch_bounds__(WAVES_PER_BLOCK * 32)
finalize_kernel(const float* __restrict__ vsum,
                const float* __restrict__ degArr,
                float* __restrict__ out, int N) {
  const int lane = threadIdx.x & 31;
  const int row  = (int)((blockIdx.x * blockDim.x + threadIdx.x) >> 5);
  if (row >= N) return;

  float4 v = ((const float4*)(vsum + (size_t)row * 128))[lane];
  float d = fmaxf(degArr[row], 1.0f);
  float s = 1.0f / d;
  v.x *= s; v.y *= s; v.z *= s; v.w *= s;

  float ss = v.x * v.x + v.y * v.y + v.z * v.z + v.w * v.w;
  #pragma unroll
  for (int m = 16; m >= 1; m >>= 1)
    ss += __shfl_xor(ss, m, 32);

  float inv = 1.0f / fmaxf(sqrtf(ss), 1e-12f);
  v.x *= inv; v.y *= inv; v.z *= inv; v.w *= inv;
  ((float4*)(out + (size_t)row * 128))[lane] = v;
}

// ---------------------------------------------------------------------------
// Zero the workspace accumulators (vsum + deg).
// ---------------------------------------------------------------------------
__global__ void zero_ws_kernel(float* __restrict__ p, size_t n) {
  size_t i = (size_t)blockIdx.x * blockDim.x + threadIdx.x;
  size_t stride = (size_t)gridDim.x * blockDim.x;
  size_t n4 = n >> 2;
  float4* p4 = (float4*)p;
  for (size_t j = i; j < n4; j += stride)
    p4[j] = make_float4(0.f, 0.f, 0.f, 0.f);
  // tail
  for (size_t j = (n4 << 2) + i; j < n; j += stride) p[j] = 0.f;
}

// ---------------------------------------------------------------------------
extern "C" void kernel_launch(void* const* d_in, const int* in_sizes, int n_in,
                              void* d_out, int out_size, void* d_ws, size_t ws_size,
                              hipStream_t stream) {
  const float*     emb  = (const float*)d_in[0];       // [N,128] f32
  const long long* mask = (const long long*)d_in[1];   // [2,E]  i64

  const int N = in_sizes[0] / 128;   // 100000
  const int E = in_sizes[1] / 2;     // 1600000

  float* vsum = (float*)d_ws;                 // [N*128] f32 accumulators
  float* deg  = vsum + (size_t)N * 128;       // [N]     f32 degree counts
  float* out  = (float*)d_out;

  // 1) zero accumulators (N*129 floats)
  size_t ztotal = (size_t)N * 129;
  zero_ws_kernel<<<2048, 256, 0, stream>>>(vsum, ztotal);

  // 2) pipelined edge scatter: 10240 waves, ~156 edges each
  const int blocks = 1280;
  const int totalWaves = blocks * WAVES_PER_BLOCK;
  scatter_kernel<<<blocks, WAVES_PER_BLOCK * 32, 0, stream>>>(
      emb, mask, vsum, deg, E, totalWaves);

  // 3) finalize: one wave per row
  const int fblocks = (N + WAVES_PER_BLOCK - 1) / WAVES_PER_BLOCK;
  finalize_kernel<<<fblocks, WAVES_PER_BLOCK * 32, 0, stream>>>(
      vsum, deg, out, N);
}